// GAT_LSTM_65231963291730
// MI455X (gfx1250) — compile-verified
//
#include <hip/hip_runtime.h>
#include <cstdint>
#include <cstddef>

// ---------------- problem constants (match reference) ----------------
#define Nn      1000     // nodes
#define NnP     1008     // nodes padded to 16-row tiles (branchless WMMA stores)
#define FIN     16       // input features
#define HDh     128      // H*D
#define NH      4        // heads
#define DH      32       // per-head dim
#define NE      16000    // edges
#define EA      (NE+Nn)  // edges + self loops = 17000
#define GB      128      // total graphs = B*T
#define CG      16       // graphs per chunk (keeps ws ~17MB, L2-resident)
#define NCHUNK  (GB/CG)
#define RTS     63       // NnP/16 row tiles
#define CTS     8        // 128/16 col tiles
#define HL      64
#define OUTF    8
#define NEG_SLOPE 0.2f

typedef __attribute__((ext_vector_type(2))) float v2f;
typedef __attribute__((ext_vector_type(8))) float v8f;

// order-preserving key for float atomic max via unsigned atomicMax
__device__ __forceinline__ unsigned fkey(float f) {
    unsigned u = __float_as_uint(f);
    return (u & 0x80000000u) ? ~u : (u | 0x80000000u);
}
__device__ __forceinline__ float funkey(unsigned k) {
    unsigned u = (k & 0x80000000u) ? (k & 0x7fffffffu) : ~k;
    return __uint_as_float(u);
}
__device__ __forceinline__ float lrelu(float v) { return v > 0.f ? v : NEG_SLOPE * v; }

__device__ __forceinline__ void edge_sd(const int* __restrict__ ei, int e, int& s, int& d) {
    if (e < NE) { s = ei[e]; d = ei[NE + e]; }
    else        { s = e - NE; d = e - NE; }        // self loop
}

__device__ __forceinline__ v2f ld2(const float* p) {
    const float2 t = *reinterpret_cast<const float2*>(p);
    v2f r; r.x = t.x; r.y = t.y; return r;
}

// ---------------- Kernel A: xp = x @ W_gat via f32 WMMA 16x16x4 ----------------
// One wave computes one 16x16 tile of xp[g]. K=16 => 4 chained WMMAs.
// B operand (W_gat, 8KB) staged in LDS once per block; A loads branchless via
// row clamp; D stores branchless into the 1008-row padded xp buffer.
__global__ __launch_bounds__(128) void k_proj_wmma(
    const float* __restrict__ x, const float* __restrict__ Wg,
    float* __restrict__ xp, int chunk)
{
    __shared__ float Wlds[FIN * HDh];                  // 8 KB of the 320 KB WGP LDS
    for (int i = threadIdx.x; i < FIN * HDh; i += 128) Wlds[i] = Wg[i];
    __syncthreads();

    const int wave   = threadIdx.x >> 5;
    const int tileId = blockIdx.x * 4 + wave;
    const int tilesPerGraph = RTS * CTS;
    if (tileId >= CG * tilesPerGraph) return;          // wave-uniform exit, EXEC stays full

    const int gl  = tileId / tilesPerGraph;
    const int rem = tileId % tilesPerGraph;
    const int rt  = rem / CTS, ct = rem % CTS;
    const int g   = chunk * CG + gl;

    const int lane = threadIdx.x & 31;
    const int half = lane >> 4;                        // 0: K+0..1 | 1: K+2..3
    const int l16  = lane & 15;

    // A fragments: lane row m = rt*16 + l16; clamp keeps loads in-bounds and
    // branchless — clamped rows only feed D rows >= Nn (padding rows in xp).
    const int m  = rt * 16 + l16;
    const int mc = (m < Nn) ? m : (Nn - 1);
    const float* arow = x + (size_t)g * (Nn * FIN) + (size_t)mc * FIN + half * 2;
    const v2f a0 = ld2(arow + 0);
    const v2f a1 = ld2(arow + 4);
    const v2f a2 = ld2(arow + 8);
    const v2f a3 = ld2(arow + 12);

    // B fragments from LDS: col = ct*16 + l16, K rows (k0 + half*2 + {0,1})
    const int cb = ct * 16 + l16;
    const int kb = half * 2;
    v2f b0, b1, b2, b3;
    b0.x = Wlds[(kb + 0) * HDh + cb];  b0.y = Wlds[(kb + 1) * HDh + cb];
    b1.x = Wlds[(kb + 4) * HDh + cb];  b1.y = Wlds[(kb + 5) * HDh + cb];
    b2.x = Wlds[(kb + 8) * HDh + cb];  b2.y = Wlds[(kb + 9) * HDh + cb];
    b3.x = Wlds[(kb + 12) * HDh + cb]; b3.y = Wlds[(kb + 13) * HDh + cb];

    v8f acc = {};
    acc = __builtin_amdgcn_wmma_f32_16x16x4_f32(false, a0, false, b0, (short)0, acc, false, false);
    acc = __builtin_amdgcn_wmma_f32_16x16x4_f32(false, a1, false, b1, (short)0, acc, false, false);
    acc = __builtin_amdgcn_wmma_f32_16x16x4_f32(false, a2, false, b2, (short)0, acc, false, false);
    acc = __builtin_amdgcn_wmma_f32_16x16x4_f32(false, a3, false, b3, (short)0, acc, false, false);

    // C/D layout: VGPR r -> row r (lanes 0-15) / row r+8 (lanes 16-31), col = lane&15.
    // xp is padded to NnP rows, so every store is in-bounds: no guards, no branches.
    float* xpg = xp + ((size_t)gl * NnP + (size_t)(rt * 16 + half * 8)) * HDh + cb;
#pragma unroll
    for (int r = 0; r < 8; ++r) {
        xpg[(size_t)r * HDh] = acc[r];
    }
}

// ---------------- Kernel B: per-node attention scores a_s, a_d ----------------
__global__ void k_att_scores(const float* __restrict__ xp,
                             const float* __restrict__ att_src,
                             const float* __restrict__ att_dst,
                             float* __restrict__ a_s, float* __restrict__ a_d)
{
    const int id = blockIdx.x * blockDim.x + threadIdx.x;
    if (id >= CG * Nn * NH) return;
    const int h  = id & 3;
    const int n  = (id >> 2) % Nn;
    const int gl = id / (Nn * NH);
    const float* row = xp + ((size_t)gl * NnP + n) * HDh + h * DH;
    const float* as  = att_src + h * DH;
    const float* ad  = att_dst + h * DH;
    float ss = 0.f, sd = 0.f;
#pragma unroll
    for (int d = 0; d < DH; ++d) { const float v = row[d]; ss += v * as[d]; sd += v * ad[d]; }
    a_s[id] = ss; a_d[id] = sd;
}

// ---------------- Kernel C: segment max over dst (pass 1) ----------------
__global__ void k_edge_max(const int* __restrict__ ei,
                           const float* __restrict__ a_s, const float* __restrict__ a_d,
                           unsigned* __restrict__ mkey)
{
    const int id = blockIdx.x * blockDim.x + threadIdx.x;
    if (id >= CG * EA) return;
    const int e = id % EA, gl = id / EA;
    int s, d; edge_sd(ei, e, s, d);
    const size_t base = (size_t)gl * Nn * NH;
#pragma unroll
    for (int h = 0; h < NH; ++h) {
        const float ev = lrelu(a_s[base + s * NH + h] + a_d[base + d * NH + h]);
        atomicMax(&mkey[base + d * NH + h], fkey(ev));
    }
}

// ---------------- Kernel D: segment sum of exp (pass 2) ----------------
__global__ void k_edge_sum(const int* __restrict__ ei,
                           const float* __restrict__ a_s, const float* __restrict__ a_d,
                           const unsigned* __restrict__ mkey, float* __restrict__ denom)
{
    const int id = blockIdx.x * blockDim.x + threadIdx.x;
    if (id >= CG * EA) return;
    const int e = id % EA, gl = id / EA;
    int s, d; edge_sd(ei, e, s, d);
    const size_t base = (size_t)gl * Nn * NH;
#pragma unroll
    for (int h = 0; h < NH; ++h) {
        const float ev = lrelu(a_s[base + s * NH + h] + a_d[base + d * NH + h]);
        const float mv = funkey(mkey[base + d * NH + h]);
        atomicAdd(&denom[base + d * NH + h], __expf(ev - mv));
    }
}

// ---------------- Kernel E: weighted aggregation (pass 3) ----------------
__global__ void k_edge_agg(const int* __restrict__ ei,
                           const float* __restrict__ a_s, const float* __restrict__ a_d,
                           const unsigned* __restrict__ mkey, const float* __restrict__ denom,
                           const float* __restrict__ xp, float* __restrict__ agg)
{
    const int id = blockIdx.x * blockDim.x + threadIdx.x;
    if (id >= CG * EA * NH) return;
    const int h  = id & 3;
    const int e  = (id >> 2) % EA;
    const int gl = id / (EA * NH);
    int s, d; edge_sd(ei, e, s, d);
    const size_t base = (size_t)gl * Nn * NH;
    const float ev = lrelu(a_s[base + s * NH + h] + a_d[base + d * NH + h]);
    const float mv = funkey(mkey[base + d * NH + h]);
    const float alpha = __expf(ev - mv) / denom[base + d * NH + h];
    const float* srow = xp  + ((size_t)gl * NnP + s) * HDh + h * DH;
    float*       drow = agg + ((size_t)gl * Nn  + d) * HDh + h * DH;
#pragma unroll 8
    for (int dd = 0; dd < DH; ++dd) atomicAdd(&drow[dd], alpha * srow[dd]);
}

// ---------------- Kernel F: bias + ReLU + global max pool ----------------
__global__ __launch_bounds__(128) void k_relu_maxpool(
    const float* __restrict__ agg, const float* __restrict__ b_gat,
    float* __restrict__ pooled, int chunk)
{
    const int gl = blockIdx.x, f = threadIdx.x;
    const float* a = agg + (size_t)gl * Nn * HDh;
    const float bias = b_gat[f];
    float best = 0.f;                                  // relu output is >= 0
    for (int n = 0; n < Nn; ++n) {
        float v = a[(size_t)n * HDh + f] + bias;
        v = v > 0.f ? v : 0.f;
        best = v > best ? v : best;
    }
    pooled[(size_t)(chunk * CG + gl) * HDh + f] = best;
}

// ---------------- Kernel G: LSTM over T steps + classifier ----------------
__global__ __launch_bounds__(256) void k_lstm_clf(
    const float* __restrict__ pooled,
    const float* __restrict__ W_ih, const float* __restrict__ W_hh,
    const float* __restrict__ b_ih, const float* __restrict__ b_hh,
    const float* __restrict__ W_clf, const float* __restrict__ b_clf,
    float* __restrict__ out)
{
    __shared__ float h[8][HL], c[8][HL], gates[8][4 * HL];
    const int tid = threadIdx.x;                       // 256 = 4*HL
    if (tid < HL)
        for (int b = 0; b < 8; ++b) { h[b][tid] = 0.f; c[b][tid] = 0.f; }
    __syncthreads();

    for (int t = 0; t < 16; ++t) {
        const int j = tid;                             // gate row 0..255
        for (int b = 0; b < 8; ++b) {
            const float* xt = pooled + (size_t)(b * 16 + t) * HDh;
            float acc = b_ih[j] + b_hh[j];
            for (int k = 0; k < HDh; ++k) acc += xt[k]   * W_ih[j * HDh + k];
            for (int k = 0; k < HL;  ++k) acc += h[b][k] * W_hh[j * HL  + k];
            gates[b][j] = acc;
        }
        __syncthreads();
        if (tid < HL) {
            const int u = tid;
            for (int b = 0; b < 8; ++b) {
                const float ig = 1.f / (1.f + __expf(-gates[b][u]));
                const float fg = 1.f / (1.f + __expf(-gates[b][HL + u]));
                const float gg = tanhf(gates[b][2 * HL + u]);
                const float og = 1.f / (1.f + __expf(-gates[b][3 * HL + u]));
                const float cc = fg * c[b][u] + ig * gg;
                c[b][u] = cc;
                h[b][u] = og * tanhf(cc);
            }
        }
        __syncthreads();
    }
    if (tid < 8 * OUTF) {
        const int b = tid >> 3, o = tid & 7;
        float acc = b_clf[o];
        for (int k = 0; k < HL; ++k) acc += h[b][k] * W_clf[o * HL + k];
        out[b * OUTF + o] = acc;
    }
}

// ---------------- host launch ----------------
extern "C" void kernel_launch(void* const* d_in, const int* in_sizes, int n_in,
                              void* d_out, int out_size, void* d_ws, size_t ws_size,
                              hipStream_t stream)
{
    (void)in_sizes; (void)n_in; (void)out_size; (void)ws_size;
    const float* x       = (const float*)d_in[0];
    const int*   ei      = (const int*)  d_in[1];
    const float* W_gat   = (const float*)d_in[2];
    const float* att_src = (const float*)d_in[3];
    const float* att_dst = (const float*)d_in[4];
    const float* b_gat   = (const float*)d_in[5];
    const float* W_ih    = (const float*)d_in[6];
    const float* W_hh    = (const float*)d_in[7];
    const float* b_ih    = (const float*)d_in[8];
    const float* b_hh    = (const float*)d_in[9];
    const float* W_clf   = (const float*)d_in[10];
    const float* b_clf   = (const float*)d_in[11];
    float* out = (float*)d_out;

    char* ws = (char*)d_ws;
    size_t off = 0;
    auto take = [&](size_t bytes) -> char* {
        char* p = ws + off;
        off = (off + bytes + 255) & ~(size_t)255;
        return p;
    };
    float*    pooled = (float*)   take((size_t)GB * HDh * sizeof(float));
    float*    xp     = (float*)   take((size_t)CG * NnP * HDh * sizeof(float));  // padded rows
    float*    agg    = (float*)   take((size_t)CG * Nn  * HDh * sizeof(float));
    float*    a_s    = (float*)   take((size_t)CG * Nn * NH * sizeof(float));
    float*    a_d    = (float*)   take((size_t)CG * Nn * NH * sizeof(float));
    unsigned* mkey   = (unsigned*)take((size_t)CG * Nn * NH * sizeof(unsigned));
    float*    denom  = (float*)   take((size_t)CG * Nn * NH * sizeof(float));

    for (int chunk = 0; chunk < NCHUNK; ++chunk) {
        hipMemsetAsync(agg,   0, (size_t)CG * Nn * HDh * sizeof(float),    stream);
        hipMemsetAsync(mkey,  0, (size_t)CG * Nn * NH  * sizeof(unsigned), stream);
        hipMemsetAsync(denom, 0, (size_t)CG * Nn * NH  * sizeof(float),    stream);

        const int tiles = CG * RTS * CTS;
        k_proj_wmma<<<(tiles + 3) / 4, 128, 0, stream>>>(x, W_gat, xp, chunk);

        const int tB = CG * Nn * NH;
        k_att_scores<<<(tB + 255) / 256, 256, 0, stream>>>(xp, att_src, att_dst, a_s, a_d);

        const int tE = CG * EA;
        k_edge_max<<<(tE + 255) / 256, 256, 0, stream>>>(ei, a_s, a_d, mkey);
        k_edge_sum<<<(tE + 255) / 256, 256, 0, stream>>>(ei, a_s, a_d, mkey, denom);

        const int tF = CG * EA * NH;
        k_edge_agg<<<(tF + 255) / 256, 256, 0, stream>>>(ei, a_s, a_d, mkey, denom, xp, agg);

        k_relu_maxpool<<<CG, 128, 0, stream>>>(agg, b_gat, pooled, chunk);
    }
    k_lstm_clf<<<1, 256, 0, stream>>>(pooled, W_ih, W_hh, b_ih, b_hh, W_clf, b_clf, out);
}